// GNN_17514876634158
// MI455X (gfx1250) — compile-verified
//
#include <hip/hip_runtime.h>

// GCN (2-layer, GCNConv with symmetric norm + self-loops) for MI455X gfx1250.
// GEMMs via V_WMMA_F32_16X16X4_F32 (exact fp32 matrix path); aggregation via
// wave-per-edge coalesced gather + global f32 atomic scatter-add.

typedef float v2f __attribute__((ext_vector_type(2)));
typedef float v8f __attribute__((ext_vector_type(8)));

// ---------------------------------------------------------------- degree
__global__ void gnn_deg_init(float* __restrict__ deg, int n) {
  int i = blockIdx.x * blockDim.x + threadIdx.x;
  if (i < n) deg[i] = 1.0f;  // self-loop contributes 1
}

__global__ void gnn_deg_accum(const int* __restrict__ dst, float* __restrict__ deg, int e) {
  int i = blockIdx.x * blockDim.x + threadIdx.x;
  if (i < e) atomicAdd(&deg[dst[i]], 1.0f);
}

__global__ void gnn_deg_rsqrt(float* __restrict__ deg, int n) {
  int i = blockIdx.x * blockDim.x + threadIdx.x;
  if (i < n) deg[i] = rsqrtf(deg[i]);  // deg -> deg^{-1/2} in place
}

// ---------------------------------------------------------------- GEMM (WMMA f32)
// H = X @ W    (X: [N,128] row-major, W: [128,NCOL] row-major)
// AGG = H * dinv[row]^2   (fused self-loop contribution / accumulator init)
// One wave computes a 16x16 tile; NCOL/16 waves per block cover all columns.
// NCOL is compile-time so all W loads use immediate offsets (no 64-bit muls).
// Requires N % 16 == 0 (N = 100000 -> 6250 row tiles).
template <int NCOL>
__global__ __launch_bounds__(256) void gnn_gemm_wmma(
    const float* __restrict__ X, const float* __restrict__ W,
    const float* __restrict__ dinv, float* __restrict__ H,
    float* __restrict__ AGG) {
  constexpr int K = 128;
  constexpr int LDSS = 132;               // pad: bank = (4*m + k) % 64, conflict-free
  __shared__ float xs[16 * LDSS];

  const int tile_m = blockIdx.x;
  const int tid = threadIdx.x;
  const int bsz = blockDim.x;

  // cooperative stage of the 16x128 X tile (float4 moves, 16B-aligned: 528%16==0)
  const float4* xg = reinterpret_cast<const float4*>(X + (size_t)tile_m * 16 * K);
  for (int idx = tid; idx < 16 * (K / 4); idx += bsz) {
    int r = idx >> 5;                     // /32
    int c4 = idx & 31;
    float4 v = xg[r * (K / 4) + c4];
    *reinterpret_cast<float4*>(&xs[r * LDSS + c4 * 4]) = v;
  }
  __syncthreads();

  const int wave = tid >> 5;
  const int lane = tid & 31;
  const int lhalf = lane >> 4;            // 0: K pair {k,k+1}; 1: {k+2,k+3}
  const int ml = lane & 15;               // A: row-in-tile, B: col-in-tile
  const int col = wave * 16 + ml;

  // base pointers folded once; unrolled-loop offsets become immediates
  const float* __restrict__ wp = W + (size_t)(2 * lhalf) * NCOL + col;
  const float* __restrict__ ap = &xs[ml * LDSS + 2 * lhalf];

  v8f c = {0.f, 0.f, 0.f, 0.f, 0.f, 0.f, 0.f, 0.f};
#pragma unroll
  for (int k = 0; k < K; k += 4) {
    // A fragment: A[m][k+2h], A[m][k+2h+1]  (8B-aligned LDS load)
    v2f a = *reinterpret_cast<const v2f*>(ap + k);
    // B fragment: B[k+2h][n], B[k+2h+1][n]  (immediate-offset b32 loads)
    v2f b;
    b.x = wp[k * NCOL];
    b.y = wp[(k + 1) * NCOL];
    c = __builtin_amdgcn_wmma_f32_16x16x4_f32(
        /*neg_a=*/false, a, /*neg_b=*/false, b,
        /*c_mod=*/(short)0, c, /*reuse_a=*/false, /*reuse_b=*/false);
  }

  // C/D layout: VGPR i -> row i (lanes 0-15) / row i+8 (lanes 16-31), col = lane&15
  const int row0 = tile_m * 16 + 8 * lhalf;
  float* __restrict__ hp = H + (size_t)row0 * NCOL + col;
  float* __restrict__ gp = AGG + (size_t)row0 * NCOL + col;
  const float* __restrict__ dp = dinv + row0;
#pragma unroll
  for (int i = 0; i < 8; ++i) {
    float h = c[i];
    float di = dp[i];
    hp[i * NCOL] = h;
    gp[i * NCOL] = h * di * di;           // self-loop msg init
  }
}

// ---------------------------------------------------------------- edge scatter-add
// One wave per edge: AGG[dst] += H[src] * dinv[src]*dinv[dst]
template <int D>
__global__ void gnn_edge_agg(const float* __restrict__ H, const float* __restrict__ dinv,
                             const int* __restrict__ src, const int* __restrict__ dst,
                             float* __restrict__ AGG, int e) {
  int gw = (blockIdx.x * blockDim.x + threadIdx.x) >> 5;
  int lane = threadIdx.x & 31;
  if (gw >= e) return;
  // speculative prefetch of the index stream ahead of this wave
  __builtin_prefetch(&src[gw + 4096], 0, 0);
  __builtin_prefetch(&dst[gw + 4096], 0, 0);
  int s = src[gw];
  int t = dst[gw];
  float norm = dinv[s] * dinv[t];
  if (D == 128) {
    const float4* hv = reinterpret_cast<const float4*>(H + (size_t)s * 128);
    float4 v = hv[lane];                              // 32 lanes x 16B = 512B coalesced
    float* op = AGG + (size_t)t * 128 + lane * 4;
    atomicAdd(op + 0, v.x * norm);
    atomicAdd(op + 1, v.y * norm);
    atomicAdd(op + 2, v.z * norm);
    atomicAdd(op + 3, v.w * norm);
  } else {  // D == 64
    const float2* hv = reinterpret_cast<const float2*>(H + (size_t)s * 64);
    float2 v = hv[lane];                              // 32 lanes x 8B = 256B
    float* op = AGG + (size_t)t * 64 + lane * 2;
    atomicAdd(op + 0, v.x * norm);
    atomicAdd(op + 1, v.y * norm);
  }
}

// ---------------------------------------------------------------- epilogues
// AGG = relu(AGG + b), D = 128 (float4-vectorized; i ranges over N*32)
__global__ void gnn_bias_relu128(float* __restrict__ AGG, const float* __restrict__ b, int n32) {
  int i = blockIdx.x * blockDim.x + threadIdx.x;
  if (i >= n32) return;
  float4 v = reinterpret_cast<float4*>(AGG)[i];
  float4 bb = reinterpret_cast<const float4*>(b)[i & 31];
  v.x = fmaxf(v.x + bb.x, 0.f);
  v.y = fmaxf(v.y + bb.y, 0.f);
  v.z = fmaxf(v.z + bb.z, 0.f);
  v.w = fmaxf(v.w + bb.w, 0.f);
  reinterpret_cast<float4*>(AGG)[i] = v;
}

// OUT += b, D = 64 (float4-vectorized; i ranges over N*16)
__global__ void gnn_bias64(float* __restrict__ OUT, const float* __restrict__ b, int n16) {
  int i = blockIdx.x * blockDim.x + threadIdx.x;
  if (i >= n16) return;
  float4 v = reinterpret_cast<float4*>(OUT)[i];
  float4 bb = reinterpret_cast<const float4*>(b)[i & 15];
  v.x += bb.x; v.y += bb.y; v.z += bb.z; v.w += bb.w;
  reinterpret_cast<float4*>(OUT)[i] = v;
}

// ---------------------------------------------------------------- driver
extern "C" void kernel_launch(void* const* d_in, const int* in_sizes, int n_in,
                              void* d_out, int out_size, void* d_ws, size_t ws_size,
                              hipStream_t stream) {
  const float* x  = (const float*)d_in[0];
  const int*   ei = (const int*)d_in[1];
  const float* W1 = (const float*)d_in[2];
  const float* b1 = (const float*)d_in[3];
  const float* W2 = (const float*)d_in[4];
  const float* b2 = (const float*)d_in[5];
  float* out = (float*)d_out;

  const int N = in_sizes[0] / 128;     // 100000
  const int E = in_sizes[1] / 2;       // 1600000
  const int* src = ei;                 // edge_index[0]
  const int* dst = ei + E;             // edge_index[1]

  // workspace: dinv[N] | h1[N*128] | agg1[N*128] | h2[N*64]  (~128.4 MB)
  float* dinv = (float*)d_ws;
  float* h1   = dinv + N;
  float* agg1 = h1 + (size_t)N * 128;
  float* h2   = agg1 + (size_t)N * 128;

  const int TB = 256;

  // degree (with self-loops) -> dinv = deg^{-1/2}
  gnn_deg_init<<<(N + TB - 1) / TB, TB, 0, stream>>>(dinv, N);
  gnn_deg_accum<<<(E + TB - 1) / TB, TB, 0, stream>>>(dst, dinv, E);
  gnn_deg_rsqrt<<<(N + TB - 1) / TB, TB, 0, stream>>>(dinv, N);

  // layer 1: h1 = x@W1 ; agg1 initialized with self-loop term h1*dinv^2
  gnn_gemm_wmma<128><<<N / 16, 256, 0, stream>>>(x, W1, dinv, h1, agg1);
  {
    long long waves = (long long)E;
    int blocks = (int)((waves * 32 + TB - 1) / TB);
    gnn_edge_agg<128><<<blocks, TB, 0, stream>>>(h1, dinv, src, dst, agg1, E);
  }
  gnn_bias_relu128<<<(N * 32 + TB - 1) / TB, TB, 0, stream>>>(agg1, b1, N * 32);

  // layer 2: h2 = relu_h@W2 ; d_out initialized with self-loop term h2*dinv^2
  gnn_gemm_wmma<64><<<N / 16, 128, 0, stream>>>(agg1, W2, dinv, h2, out);
  {
    long long waves = (long long)E;
    int blocks = (int)((waves * 32 + TB - 1) / TB);
    gnn_edge_agg<64><<<blocks, TB, 0, stream>>>(h2, dinv, src, dst, out, E);
  }
  gnn_bias64<<<(N * 16 + TB - 1) / TB, TB, 0, stream>>>(out, b2, N * 16);
}